// TypedLinear_55705725829586
// MI455X (gfx1250) — compile-verified
//
#include <hip/hip_runtime.h>

#define NTOK   131072
#define NTYPE  8
#define INF    128
#define OUTF   128
#define MTILE  64
#define SENT   0xFFFFFFFFu
#define PADW   132   // padded LDS row stride (dwords): 132 % 64 = 4 -> conflict-free b64 reads

typedef __attribute__((ext_vector_type(2))) float v2f;
typedef __attribute__((ext_vector_type(4))) float v4f;
typedef __attribute__((ext_vector_type(8))) float v8f;

// d_ws layout (uints):
//   [0..8)    counts
//   [8..17)   poff  (padded segment offsets, poff[8] = padded total)
//   [24..32)  cursor
//   [32..)    idx   (NTOK + 512 entries)  -> needs ~515 KB of workspace

__global__ void tl_zero(unsigned* ws) {
    if (threadIdx.x < NTYPE) ws[threadIdx.x] = 0u;
}

__global__ void tl_hist(const int* __restrict__ types, unsigned* __restrict__ ws) {
    __shared__ unsigned lcnt[NTYPE];
    int tid = threadIdx.x;
    if (tid < NTYPE) lcnt[tid] = 0u;
    __syncthreads();
    int i = blockIdx.x * blockDim.x + tid;
    if (i < NTOK) atomicAdd(&lcnt[((unsigned)types[i]) & 7u], 1u);
    __syncthreads();
    if (tid < NTYPE && lcnt[tid]) atomicAdd(&ws[tid], lcnt[tid]);
}

__global__ void tl_scan(unsigned* ws) {
    if (threadIdx.x == 0) {
        unsigned* counts = ws;
        unsigned* poff   = ws + 8;
        unsigned* cursor = ws + 24;
        unsigned* idx    = ws + 32;
        unsigned off = 0;
        for (int t = 0; t < NTYPE; ++t) {
            poff[t]   = off;
            cursor[t] = off;
            unsigned c  = counts[t];
            unsigned pc = (c + (MTILE - 1u)) & ~(MTILE - 1u);  // pad segment to M-tile
            for (unsigned j = c; j < pc; ++j) idx[off + j] = SENT;
            off += pc;
        }
        poff[NTYPE] = off;
    }
}

__global__ void tl_scatter(const int* __restrict__ types, unsigned* __restrict__ ws) {
    unsigned* cursor = ws + 24;
    unsigned* idx    = ws + 32;
    __shared__ unsigned lcnt[NTYPE];
    __shared__ unsigned lbase[NTYPE];
    int tid = threadIdx.x;
    if (tid < NTYPE) lcnt[tid] = 0u;
    __syncthreads();
    int i = blockIdx.x * blockDim.x + tid;
    unsigned t = 0, r = 0;
    bool valid = (i < NTOK);
    if (valid) {
        t = ((unsigned)types[i]) & 7u;
        r = atomicAdd(&lcnt[t], 1u);
    }
    __syncthreads();
    if (tid < NTYPE && lcnt[tid]) lbase[tid] = atomicAdd(&cursor[tid], lcnt[tid]);
    __syncthreads();
    if (valid) idx[lbase[t] + r] = (unsigned)i;
}

__device__ __forceinline__ v8f wmma_f32(v2f a, v2f b, v8f c) {
    // V_WMMA_F32_16X16X4_F32 : D = A(16x4) * B(4x16) + C(16x16), fp32
    return __builtin_amdgcn_wmma_f32_16x16x4_f32(false, a, false, b, (short)0, c, false, false);
}

__global__ void __launch_bounds__(256) tl_gemm(const float* __restrict__ x,
                                               const float* __restrict__ W,
                                               const float* __restrict__ bias,
                                               float* __restrict__ out,
                                               const unsigned* __restrict__ ws) {
    extern __shared__ float smem[];
    float*    w_lds = smem;                       // 128 x PADW
    float*    x_lds = smem + OUTF * PADW;         // 64  x PADW
    unsigned* s_idx = (unsigned*)(smem + OUTF * PADW + MTILE * PADW); // 64

    const unsigned* poff = ws + 8;
    const unsigned* idx  = ws + 32;

    const unsigned tile_base = blockIdx.x * (unsigned)MTILE;
    const unsigned total     = poff[NTYPE];
    if (tile_base >= total) return;                 // uniform exit

    // segment -> type (segments are MTILE-aligned, so a tile never straddles types)
    unsigned t = 0;
    #pragma unroll
    for (unsigned k = 1; k < NTYPE; ++k)
        if (tile_base >= poff[k]) t = k;

    const int tid = threadIdx.x;
    if (tid < MTILE) s_idx[tid] = idx[tile_base + tid];
    __syncthreads();

    // Gather 64 x-rows into LDS: 2048 float4 / 256 threads = 8 each
    #pragma unroll
    for (int r = 0; r < 8; ++r) {
        int flat = r * 256 + tid;
        int row  = flat >> 5;          // 0..63
        int col  = (flat & 31) * 4;    // 0..124
        unsigned tok = s_idx[row];
        v4f v = {0.f, 0.f, 0.f, 0.f};
        if (tok != SENT) v = *(const v4f*)(x + (size_t)tok * INF + col);
        *(v4f*)(x_lds + row * PADW + col) = v;
    }
    // Stage W[t] (64 KB) into LDS: 4096 float4 / 256 threads = 16 each
    const float* Wt = W + (size_t)t * OUTF * INF;
    #pragma unroll
    for (int r = 0; r < 16; ++r) {
        int flat = r * 256 + tid;
        int row  = flat >> 5;          // 0..127 (output feature n)
        int col  = (flat & 31) * 4;    // 0..124 (k)
        v4f v = *(const v4f*)(Wt + row * INF + col);
        *(v4f*)(w_lds + row * PADW + col) = v;
    }
    __syncthreads();

    // 8 waves in a 2(M) x 4(N) grid; each wave owns a 32x32 patch = 2x2 WMMA tiles
    const int wave = tid >> 5;
    const int lane = tid & 31;
    const int hi   = lane >> 4;        // half-wave select
    const int l    = lane & 15;
    const int m0   = (wave >> 2) * 32;
    const int n0   = (wave & 3) * 32;

    // A fragment: lanes 0-15 carry K=k,k+1 ; lanes 16-31 carry K=k+2,k+3 (ISA 16x4 layout)
    const float* pa0 = x_lds + (m0 + l)      * PADW + 2 * hi;
    const float* pa1 = x_lds + (m0 + 16 + l) * PADW + 2 * hi;
    const float* pb0 = w_lds + (n0 + l)      * PADW + 2 * hi;
    const float* pb1 = w_lds + (n0 + 16 + l) * PADW + 2 * hi;

    v8f acc00 = {}; v8f acc01 = {}; v8f acc10 = {}; v8f acc11 = {};

    #pragma unroll 4
    for (int k = 0; k < INF; k += 4) {
        v2f a0 = *(const v2f*)(pa0 + k);
        v2f a1 = *(const v2f*)(pa1 + k);
        v2f b0 = *(const v2f*)(pb0 + k);
        v2f b1 = *(const v2f*)(pb1 + k);
        acc00 = wmma_f32(a0, b0, acc00);
        acc01 = wmma_f32(a0, b1, acc01);
        acc10 = wmma_f32(a1, b0, acc10);
        acc11 = wmma_f32(a1, b1, acc11);
    }

    const float bias0 = bias[t * OUTF + n0 + l];
    const float bias1 = bias[t * OUTF + n0 + 16 + l];

    // C/D layout: VGPR v, lanes 0-15 -> M=v ; lanes 16-31 -> M=8+v ; N = n + (lane&15)
    #pragma unroll
    for (int v = 0; v < 8; ++v) {
        int mr0 = m0 + v + 8 * hi;
        int mr1 = m0 + 16 + v + 8 * hi;
        unsigned tok0 = s_idx[mr0];
        unsigned tok1 = s_idx[mr1];
        if (tok0 != SENT) {
            out[(size_t)tok0 * OUTF + n0 + l]      = acc00[v] + bias0;
            out[(size_t)tok0 * OUTF + n0 + 16 + l] = acc01[v] + bias1;
        }
        if (tok1 != SENT) {
            out[(size_t)tok1 * OUTF + n0 + l]      = acc10[v] + bias0;
            out[(size_t)tok1 * OUTF + n0 + 16 + l] = acc11[v] + bias1;
        }
    }
}

extern "C" void kernel_launch(void* const* d_in, const int* in_sizes, int n_in,
                              void* d_out, int out_size, void* d_ws, size_t ws_size,
                              hipStream_t stream) {
    (void)in_sizes; (void)n_in; (void)out_size; (void)ws_size;
    const float* x     = (const float*)d_in[0];
    const int*   types = (const int*)d_in[1];
    const float* W     = (const float*)d_in[2];
    const float* bias  = (const float*)d_in[3];
    float*       out   = (float*)d_out;
    unsigned*    ws    = (unsigned*)d_ws;

    tl_zero   <<<1, 32, 0, stream>>>(ws);
    tl_hist   <<<NTOK / 256, 256, 0, stream>>>(types, ws);
    tl_scan   <<<1, 32, 0, stream>>>(ws);
    tl_scatter<<<NTOK / 256, 256, 0, stream>>>(types, ws);

    size_t smem = (size_t)(OUTF * PADW + MTILE * PADW) * sizeof(float) + MTILE * sizeof(unsigned);
    hipFuncSetAttribute((const void*)tl_gemm,
                        hipFuncAttributeMaxDynamicSharedMemorySize, (int)smem);
    tl_gemm<<<NTOK / MTILE + NTYPE, 256, smem, stream>>>(x, W, bias, out, ws);
}